// SparseArchTraceAbleWrapper_62620623176140
// MI455X (gfx1250) — compile-verified
//
#include <hip/hip_runtime.h>
#include <stdint.h>

// Problem constants (from the reference)
constexpr int kF = 26;
constexpr int kV = 100000;
constexpr int kD = 128;          // embedding dim: 128 f32 = 512 B = 32 lanes x 16 B
constexpr int kB = 4096;
constexpr int kN = kB * 20;      // indices per table

constexpr int kWaves = 8;        // waves per block (256 threads)
constexpr int kNBuf  = 8;        // LDS row slots per wave (batch of async gathers)

// The builtin wants typed int4 pointers: (AS1 int4* global_src, AS3 int4* lds_dst, imm, imm)
typedef int v4i __attribute__((ext_vector_type(4)));
typedef __attribute__((address_space(1))) v4i* GlobalV4iPtr;
typedef __attribute__((address_space(3))) v4i* LdsV4iPtr;
typedef __attribute__((address_space(3))) void LdsVoid;

// One CDNA5 async gather: 32 lanes x b128 = one 512B embedding row -> LDS.
__device__ __forceinline__ void row_gather_async(const float* gsrc, float* ldst) {
#if __has_builtin(__builtin_amdgcn_global_load_async_to_lds_b128)
  __builtin_amdgcn_global_load_async_to_lds_b128(
      (GlobalV4iPtr)(void*)gsrc, (LdsV4iPtr)(void*)ldst, /*offset=*/0, /*cpol=*/0);
#else
  unsigned laddr = (unsigned)(uintptr_t)(LdsVoid*)ldst;  // 32-bit LDS byte address
  asm volatile("global_load_async_to_lds_b128 %0, %1, off"
               :: "v"(laddr), "v"(gsrc)
               : "memory");
#endif
}

__device__ __forceinline__ void wait_async_zero() {
#if __has_builtin(__builtin_amdgcn_s_wait_asynccnt)
  __builtin_amdgcn_s_wait_asynccnt(0);
  asm volatile("" ::: "memory");   // compiler fence: LDS now holds fresh rows
#else
  asm volatile("s_wait_asynccnt 0" ::: "memory");
#endif
}

__device__ __forceinline__ void wait_ds_zero() {
  // Ensure previous batch's LDS reads have retired before the async engine
  // overwrites the slots.
  asm volatile("s_wait_dscnt 0" ::: "memory");
}

__global__ __launch_bounds__(kWaves * 32)
void embag_pool_kernel(const int* __restrict__ values,    // [F, N]
                       const int* __restrict__ offsets,   // [F, B+1]
                       const float* __restrict__ tables,  // [F, V, D]
                       float* __restrict__ out) {         // [B, F, D]
  __shared__ __align__(16) float smem[kWaves * kNBuf * kD];  // 32 KB / block

  const int lane = threadIdx.x & 31;
  const int wave = threadIdx.x >> 5;

  // Scalarize the (table, bag) id so offset/index fetches go down the SMEM
  // (s_load / KMcnt) path instead of broadcast vector loads.
  const int pair = __builtin_amdgcn_readfirstlane((int)(blockIdx.x * kWaves + wave));
  const int f = pair / kB;   // consecutive waves share a table (index locality)
  const int b = pair % kB;

  const int*   offs = offsets + f * (kB + 1);
  const int    p0   = offs[b];
  const int    p1   = offs[b + 1];
  const int*   vals = values + (size_t)f * kN;
  const float* tab  = tables + (size_t)f * kV * kD;

  float* slots = smem + wave * (kNBuf * kD);
  const int laneOff = lane * 4;  // 4 floats (16 B) per lane

  float4 acc = make_float4(0.f, 0.f, 0.f, 0.f);

  for (int p = p0; p < p1; p += kNBuf) {
    const int rem = p1 - p;
    const int cnt = rem < kNBuf ? rem : kNBuf;

    wait_ds_zero();  // slots free for reuse
#pragma unroll
    for (int j = 0; j < kNBuf; ++j) {
      if (j < cnt) {
        const int idx = vals[p + j];                       // scalar load
        const float* row = tab + (size_t)idx * kD + laneOff;
        row_gather_async(row, slots + j * kD + laneOff);   // 512 B row -> LDS
      }
    }
    wait_async_zero();  // ASYNCcnt == 0: all cnt rows resident in LDS

#pragma unroll
    for (int j = 0; j < kNBuf; ++j) {
      if (j < cnt) {
        const float4 v = *(const float4*)(slots + j * kD + laneOff);  // ds_load_b128
        acc.x += v.x; acc.y += v.y; acc.z += v.z; acc.w += v.w;
      }
    }
  }

  // Coalesced 512 B store: out[b][f][0:128]
  float4* dst = (float4*)(out + ((size_t)b * kF + f) * kD) + lane;
  *dst = acc;
}

extern "C" void kernel_launch(void* const* d_in, const int* in_sizes, int n_in,
                              void* d_out, int out_size, void* d_ws, size_t ws_size,
                              hipStream_t stream) {
  (void)in_sizes; (void)n_in; (void)out_size; (void)d_ws; (void)ws_size;
  const int*   values  = (const int*)d_in[0];
  const int*   offsets = (const int*)d_in[1];
  const float* tables  = (const float*)d_in[2];
  float*       out     = (float*)d_out;

  const int totalPairs = kF * kB;                 // 106,496 waves
  dim3 grid(totalPairs / kWaves);                 // 13,312 blocks
  dim3 block(kWaves * 32);                        // 256 threads = 8 wave32
  hipLaunchKernelGGL(embag_pool_kernel, grid, block, 0, stream,
                     values, offsets, tables, out);
}